// PushUp_67181878444254
// MI455X (gfx1250) — compile-verified
//
#include <hip/hip_runtime.h>
#include <hip/hip_bf16.h>

// Problem shape (structural constants from the reference):
//   features     : (N, 128)  f32
//   weights_down : (N, 5)    f32
//   nidx_down    : (N, 5)    i32
//   sel_idx_up   : (N_UP, 1) i32
//   out          : (N_UP, 128) f32
#define FDIM   128
#define KN     5
#define ROWSTR 132   // 128 features + wsum @128 + 3 pad -> 528B rows, 16B aligned

// ---------------------------------------------------------------------------
// Zero the accumulator (pushed) and the destination mask. b128 stores.
// ---------------------------------------------------------------------------
__global__ void zero_ws_kernel(float* __restrict__ pushed, size_t nfloat,
                               unsigned char* __restrict__ mask, size_t nmask) {
    size_t tid    = (size_t)blockIdx.x * blockDim.x + threadIdx.x;
    size_t stride = (size_t)gridDim.x * blockDim.x;
    size_t nvec   = nfloat >> 2;
    float4 z = make_float4(0.f, 0.f, 0.f, 0.f);
    for (size_t i = tid; i < nvec; i += stride)
        ((float4*)pushed)[i] = z;
    // tail (nfloat may not be a multiple of 4 for the fallback stride)
    for (size_t i = (nvec << 2) + tid; i < nfloat; i += stride)
        pushed[i] = 0.f;
    if (mask) {
        for (size_t i = tid; i < nmask; i += stride)
            mask[i] = 0;
    }
}

// ---------------------------------------------------------------------------
// Mark destination rows that will actually be gathered (unique ~22% of N).
// ---------------------------------------------------------------------------
__global__ void mark_kernel(const int* __restrict__ sel,
                            unsigned char* __restrict__ mask,
                            int nup, int n) {
    int i = blockIdx.x * blockDim.x + threadIdx.x;
    if (i < nup) {
        int s = sel[i];
        if (s >= 0 && s < n) mask[s] = 1;
    }
}

// ---------------------------------------------------------------------------
// Scatter: one 128-thread block per source node. Thread t owns feature t.
// For each of the K=5 neighbors: skip if destination row is never read
// (mask == 0), else 128 coalesced global_atomic_add_f32 plus one for wsum.
// nidx/weights loads are blockIdx-uniform -> scalarized by the compiler.
// ---------------------------------------------------------------------------
__global__ __launch_bounds__(FDIM) void
scatter_kernel(const float* __restrict__ features,
               const float* __restrict__ wdown,
               const int*   __restrict__ nidx,
               const unsigned char* __restrict__ mask,
               float* __restrict__ pushed,
               int n, int rowstr) {
    int node = blockIdx.x;
    int t    = threadIdx.x;

    float fv = features[(size_t)node * FDIM + t];

    // gfx1250 prefetch of the next node's feature row (global_prefetch_b8)
    if (node + 1 < n)
        __builtin_prefetch(&features[(size_t)(node + 1) * FDIM + t], 0, 3);

#pragma unroll
    for (int k = 0; k < KN; ++k) {
        int idx = nidx[(size_t)node * KN + k];
        if (idx < 0) continue;                    // reference validity mask
        if (mask && mask[idx] == 0) continue;     // row never gathered: skip
        float wk = wdown[(size_t)node * KN + k];
        float* row = pushed + (size_t)idx * rowstr;
        unsafeAtomicAdd(row + t, wk * fv);        // global_atomic_add_f32
        if (t == 0)
            unsafeAtomicAdd(row + FDIM, wk);      // wsum column
    }
}

// ---------------------------------------------------------------------------
// Gather + normalize: one 128-thread block per selected row.
//   wsum = relu(w); wsum = (wsum > 0) ? wsum : 1e-3; out = feat / wsum
// ---------------------------------------------------------------------------
__global__ __launch_bounds__(FDIM) void
gather_kernel(const float* __restrict__ pushed,
              const int*   __restrict__ sel,
              float*       __restrict__ out,
              int nup, int rowstr) {
    int u = blockIdx.x;
    if (u >= nup) return;
    int t   = threadIdx.x;
    int idx = sel[u];

    const float* row = pushed + (size_t)idx * rowstr;
    float w = row[FDIM];                 // uniform address -> single request
    float wsum = (w > 0.f) ? w : 0.001f; // relu + floor, matches reference
    float v = row[t];
    out[(size_t)u * FDIM + t] = v / wsum;
}

// ---------------------------------------------------------------------------
// Host-side launcher
// ---------------------------------------------------------------------------
extern "C" void kernel_launch(void* const* d_in, const int* in_sizes, int n_in,
                              void* d_out, int out_size, void* d_ws, size_t ws_size,
                              hipStream_t stream) {
    const float* features = (const float*)d_in[0];
    const float* wdown    = (const float*)d_in[1];
    const int*   nidx     = (const int*)d_in[2];
    const int*   sel      = (const int*)d_in[3];
    float*       out      = (float*)d_out;

    const int N   = in_sizes[0] / FDIM;   // 400000
    const int NUP = in_sizes[3];          // 100000

    // Workspace layout: [ pushed : N*rowstr floats ][ mask : N bytes ]
    // Prefer the 16B-aligned stride 132; fall back to 129 if workspace tight.
    int rowstr = ROWSTR;
    size_t pushed_bytes = (size_t)N * rowstr * sizeof(float);
    if (ws_size < pushed_bytes + (size_t)N) {
        rowstr = FDIM + 1;                // 129, minimal
        pushed_bytes = (size_t)N * rowstr * sizeof(float);
    }
    float* pushed = (float*)d_ws;
    unsigned char* mask = nullptr;
    if (ws_size >= pushed_bytes + (size_t)N)
        mask = (unsigned char*)d_ws + pushed_bytes;

    const size_t nfloat = (size_t)N * rowstr;

    // 1) zero accumulator + mask
    zero_ws_kernel<<<2048, 256, 0, stream>>>(pushed, nfloat, mask, (size_t)N);

    // 2) mark rows that will be gathered
    if (mask)
        mark_kernel<<<(NUP + 255) / 256, 256, 0, stream>>>(sel, mask, NUP, N);

    // 3) masked scatter-add (dominant phase; ~78% of atomics eliminated)
    scatter_kernel<<<N, FDIM, 0, stream>>>(features, wdown, nidx, mask,
                                           pushed, N, rowstr);

    // 4) gather + normalize
    gather_kernel<<<NUP, FDIM, 0, stream>>>(pushed, sel, out, NUP, rowstr);
}